// Matcher_58695023067403
// MI455X (gfx1250) — compile-verified
//
#include <hip/hip_runtime.h>
#include <hip/hip_bf16.h>

typedef __attribute__((ext_vector_type(2))) float v2f;
typedef __attribute__((ext_vector_type(8))) float v8f;

#define DDIM 256
#define BM 128
#define BN 128
#define LDT 260  // padded LDS row stride (floats): 260 mod 64 = 4 -> conflict-free frags

// ---------------------------------------------------------------------------
// Kernel 1: inverse clamped norms  inv[src][w][p][i] = 1/max(||x_i * W_w[p]||, eps)
// src: 0 = sentence1, 1 = sentence2.  Table: [2][2][32][1024] floats in d_ws.
// ---------------------------------------------------------------------------
__global__ __launch_bounds__(256) void norms_kernel(
    const float* __restrict__ s1, const float* __restrict__ s2,
    const float* __restrict__ W1, const float* __restrict__ W2,
    float* __restrict__ inv)
{
    int idx = blockIdx.x * blockDim.x + threadIdx.x;   // 131072 total
    int i   = idx & 1023;
    int p   = (idx >> 10) & 31;
    int w   = (idx >> 15) & 1;
    int src = (idx >> 16) & 1;
    const float4* xr = (const float4*)((src ? s2 : s1) + i * DDIM);
    const float4* wr = (const float4*)((w ? W2 : W1) + p * DDIM);
    float sum = 0.f;
#pragma unroll 8
    for (int d4 = 0; d4 < DDIM / 4; ++d4) {
        float4 xv = xr[d4];
        float4 wv = wr[d4];
        float a0 = xv.x * wv.x, a1 = xv.y * wv.y, a2 = xv.z * wv.z, a3 = xv.w * wv.w;
        sum += a0 * a0 + a1 * a1 + a2 * a2 + a3 * a3;
    }
    inv[idx] = 1.0f / fmaxf(sqrtf(sum), 1e-8f);
}

// ---------------------------------------------------------------------------
// Kernel 2: per (w,p): rowmax_j of (s1_hat @ s2_hat^T) via V_WMMA_F32_16X16X4_F32
// grid.x = 64 pslots * 8 i-tiles; block = 256 threads = 8 waves.
// Wave -> 32x64 C sub-block (2 row tiles x 4 col tiles): per K=4 step,
// 6 ds_b64 fragment loads feed 8 WMMAs (vs 9:8 for a 16x128 strip).
// ---------------------------------------------------------------------------
__global__ __launch_bounds__(256) void mpm_kernel(
    const float* __restrict__ s1, const float* __restrict__ s2,
    const float* __restrict__ W1, const float* __restrict__ W2,
    const float* __restrict__ invn,   // [2][2][32][1024]
    float* __restrict__ out)          // [2][1024][32] (fwd then bwd)
{
    __shared__ float lds_W[DDIM];
    __shared__ float lds_A[BM * LDT];
    __shared__ float lds_B[BN * LDT];
    __shared__ float red[2 * BM];     // [colBlk][localRow] partial row-maxes

    const int pslot = blockIdx.x & 63;   // w*32 + p
    const int iTile = blockIdx.x >> 6;   // 0..7
    const int w  = pslot >> 5;
    const int p  = pslot & 31;
    const int i0 = iTile * BM;

    const float* Wsel = w ? W2 : W1;
    const float* invA = invn + ((size_t)(0 * 2 + w) * 32 + p) * 1024;
    const float* invB = invn + ((size_t)(1 * 2 + w) * 32 + p) * 1024;

    const int t    = threadIdx.x;
    const int lane = t & 31;
    const int wv   = t >> 5;            // wave id 0..7
    const int rowBlk = wv & 3;          // 4 row blocks of 32
    const int colBlk = wv >> 2;         // 2 col blocks of 64

    // Stage perspective weight row.
    if (t < DDIM / 4)
        ((float4*)lds_W)[t] = ((const float4*)(Wsel + p * DDIM))[t];
    __syncthreads();

    // Stage A tile: A_hat[i,d] = s1[i0+i,d] * W[p,d] * invA[i0+i]
#pragma unroll 8
    for (int it = 0; it < (BM * DDIM / 4) / 256; ++it) {
        int slot = t + it * 256;        // 8192 float4 slots
        int row  = slot >> 6;           // /(DDIM/4)
        int c4   = slot & 63;
        float ir = invA[i0 + row];
        float4 v  = ((const float4*)(s1 + (size_t)(i0 + row) * DDIM))[c4];
        float4 wq = ((const float4*)lds_W)[c4];
        v.x *= wq.x * ir; v.y *= wq.y * ir; v.z *= wq.z * ir; v.w *= wq.w * ir;
        *((float4*)&lds_A[row * LDT + c4 * 4]) = v;
    }

    float rmax[2][8];
#pragma unroll
    for (int rt = 0; rt < 2; ++rt)
#pragma unroll
        for (int r = 0; r < 8; ++r) rmax[rt][r] = -3.402823466e+38f;

    const int m     = lane & 15;        // row (A) / col (B) within 16x16 tile
    const int khalf = lane >> 4;        // 0: k,k+1   1: k+2,k+3
    const float* Arow0 = &lds_A[(rowBlk * 32 + m) * LDT];
    const float* Arow1 = Arow0 + 16 * LDT;

    for (int jt = 0; jt < 8; ++jt) {
        const int j0 = jt * BN;
        __syncthreads();                // previous lds_B readers done
#pragma unroll 8
        for (int it = 0; it < (BN * DDIM / 4) / 256; ++it) {
            int slot = t + it * 256;
            int row  = slot >> 6;
            int c4   = slot & 63;
            float ir = invB[j0 + row];
            float4 v  = ((const float4*)(s2 + (size_t)(j0 + row) * DDIM))[c4];
            float4 wq = ((const float4*)lds_W)[c4];
            v.x *= wq.x * ir; v.y *= wq.y * ir; v.z *= wq.z * ir; v.w *= wq.w * ir;
            *((float4*)&lds_B[row * LDT + c4 * 4]) = v;
        }
        __syncthreads();

        v8f acc0[4], acc1[4];
#pragma unroll
        for (int nt = 0; nt < 4; ++nt) {
            acc0[nt] = (v8f){0.f,0.f,0.f,0.f,0.f,0.f,0.f,0.f};
            acc1[nt] = (v8f){0.f,0.f,0.f,0.f,0.f,0.f,0.f,0.f};
        }

#pragma unroll 4
        for (int k = 0; k < DDIM; k += 4) {
            const int kk = k + 2 * khalf;
            v2f a0 = *((const v2f*)(Arow0 + kk));
            v2f a1 = *((const v2f*)(Arow1 + kk));
#pragma unroll
            for (int nt = 0; nt < 4; ++nt) {
                v2f b = *((const v2f*)(&lds_B[(colBlk * 64 + nt * 16 + m) * LDT] + kk));
                acc0[nt] = __builtin_amdgcn_wmma_f32_16x16x4_f32(
                    false, a0, false, b, (short)0, acc0[nt], false, false);
                acc1[nt] = __builtin_amdgcn_wmma_f32_16x16x4_f32(
                    false, a1, false, b, (short)0, acc1[nt], false, false);
            }
        }

        // fold C tiles into running max over j
#pragma unroll
        for (int nt = 0; nt < 4; ++nt) {
#pragma unroll
            for (int r = 0; r < 8; ++r) {
                rmax[0][r] = fmaxf(rmax[0][r], acc0[nt][r]);
                rmax[1][r] = fmaxf(rmax[1][r], acc1[nt][r]);
            }
        }
    }

    // Cross-lane max over the 16 columns held in each half-wave group.
#pragma unroll
    for (int rt = 0; rt < 2; ++rt) {
#pragma unroll
        for (int r = 0; r < 8; ++r) {
            float v = rmax[rt][r];
            v = fmaxf(v, __shfl_xor(v, 1, 32));
            v = fmaxf(v, __shfl_xor(v, 2, 32));
            v = fmaxf(v, __shfl_xor(v, 4, 32));
            v = fmaxf(v, __shfl_xor(v, 8, 32));
            rmax[rt][r] = v;
        }
    }

    // Stage per-wave partial row maxima (over this wave's 64 columns) to LDS.
    const int sel = lane & 7;
    const int grp = lane >> 4;          // 0: rows 0..7 of tile, 1: rows 8..15
#pragma unroll
    for (int rt = 0; rt < 2; ++rt) {
        float v = rmax[rt][0];
#pragma unroll
        for (int r = 1; r < 8; ++r) v = (sel == r) ? rmax[rt][r] : v;
        if ((lane & 15) < 8) {
            int rowLocal = rowBlk * 32 + rt * 16 + grp * 8 + sel;
            red[colBlk * BM + rowLocal] = v;
        }
    }
    __syncthreads();

    // Combine the two column-halves and store the final [row, p] results.
    if (t < BM) {
        int row = i0 + t;
        out[(size_t)w * 1024 * 32 + (size_t)row * 32 + p] =
            fmaxf(red[t], red[BM + t]);
    }
}

extern "C" void kernel_launch(void* const* d_in, const int* in_sizes, int n_in,
                              void* d_out, int out_size, void* d_ws, size_t ws_size,
                              hipStream_t stream) {
    const float* s1 = (const float*)d_in[0];   // [1024,256]
    const float* s2 = (const float*)d_in[1];   // [1024,256]
    const float* W1 = (const float*)d_in[2];   // [32,256]
    const float* W2 = (const float*)d_in[3];   // [32,256]
    float* out = (float*)d_out;                // [2][1024][32]
    float* inv = (float*)d_ws;                 // [2][2][32][1024] = 512 KB

    norms_kernel<<<dim3(512), dim3(256), 0, stream>>>(s1, s2, W1, W2, inv);
    mpm_kernel<<<dim3(512), dim3(256), 0, stream>>>(s1, s2, W1, W2, inv, out);
}